// DTSHLoss_35940286333614
// MI455X (gfx1250) — compile-verified
//
#include <hip/hip_runtime.h>
#include <hip/hip_bf16.h>

#define ALPHA 5.0f
#define LMBD  1.0f
#define B     512
#define NBIT  64
#define NCLASS 100

typedef __attribute__((ext_vector_type(2))) float v2f;
typedef __attribute__((ext_vector_type(8))) float v8f;

// ---------------------------------------------------------------------------
// Kernel 0: zero the 3 accumulators (rowmean sum, valid count, quant-sse sum)
// ---------------------------------------------------------------------------
__global__ void init_accum_kernel(float* accum) {
    int i = threadIdx.x;
    if (i < 3) accum[i] = 0.0f;
}

// ---------------------------------------------------------------------------
// Kernel 1: labels[i] = argmax_c y[i,c]  (y is one-hot)
// ---------------------------------------------------------------------------
__global__ void labels_kernel(const float* __restrict__ y, int* __restrict__ labels) {
    int i = blockIdx.x * blockDim.x + threadIdx.x;
    if (i < B) {
        int lab = 0;
        for (int c = 0; c < NCLASS; ++c) {
            if (y[i * NCLASS + c] > 0.5f) { lab = c; break; }
        }
        labels[i] = lab;
    }
}

// ---------------------------------------------------------------------------
// Kernel 2: ip = u @ u^T via V_WMMA_F32_16X16X4_F32.
// One wave (32 threads) per 16x16 output tile; K=64 in 16 steps of 4.
// A 16x4 layout: lane L -> row M=L&15; VGPR0/1 = K {0,1} (lanes 0-15) or
// K {2,3} (lanes 16-31). B 4x16 is the transposed row-block of u, which by
// symmetry uses the identical per-lane indexing with the N tile's rows.
// ---------------------------------------------------------------------------
__global__ void __launch_bounds__(32)
gemm_ip_wmma_kernel(const float* __restrict__ u, float* __restrict__ ip) {
    const int tm   = blockIdx.x;          // row-tile (M)
    const int tn   = blockIdx.y;          // col-tile (N)
    const int lane = threadIdx.x;         // 0..31, EXEC all ones (required)
    const int r     = lane & 15;
    const int halfk = (lane >> 4) << 1;   // 0 for lanes 0-15, 2 for lanes 16-31

    const float* __restrict__ arow = u + (tm * 16 + r) * NBIT;  // A: rows of u
    const float* __restrict__ brow = u + (tn * 16 + r) * NBIT;  // B: rows of u (transposed use)

    v8f acc = {};
#pragma unroll
    for (int kk = 0; kk < NBIT; kk += 4) {
        if (kk + 8 < NBIT) {
            __builtin_prefetch(arow + kk + 8, 0, 3);   // global_prefetch_b8
            __builtin_prefetch(brow + kk + 8, 0, 3);
        }
        v2f a, b;
        a.x = arow[kk + halfk];
        a.y = arow[kk + halfk + 1];
        b.x = brow[kk + halfk];
        b.y = brow[kk + halfk + 1];
        // (neg_a, A, neg_b, B, c_mod, C, reuse_a, reuse_b)
        acc = __builtin_amdgcn_wmma_f32_16x16x4_f32(
            false, a, false, b, (short)0, acc, false, false);
    }

    // D layout: acc[v] at lane L -> M = v + 8*(L>>4), N = L&15
    const int mbase = tm * 16 + ((lane >> 4) << 3);
    const int n     = tn * 16 + (lane & 15);
#pragma unroll
    for (int v = 0; v < 8; ++v) {
        ip[(mbase + v) * B + n] = acc[v];
    }
}

// ---------------------------------------------------------------------------
// Kernel 3: per-anchor-row triplet loss. One 256-thread block per row b.
// Compact positive ip-values into LDS (n_pos ~ B/NCLASS, small), then each
// thread sweeps its strided negatives against all positives.
// ---------------------------------------------------------------------------
__global__ void __launch_bounds__(256)
triplet_row_kernel(const float* __restrict__ ip, const int* __restrict__ labels,
                   float* __restrict__ accum) {
    const int b   = blockIdx.x;
    const int tid = threadIdx.x;
    const int lb  = labels[b];

    __shared__ int   s_cnt;
    __shared__ float s_posval[B];
    __shared__ float s_red[256];

    if (tid == 0) s_cnt = 0;
    __syncthreads();

    const float* __restrict__ iprow = ip + (size_t)b * B;

    // compact positives (diagonal included, matching the reference mask)
    for (int j = tid; j < B; j += 256) {
        if (labels[j] == lb) {
            int idx = atomicAdd(&s_cnt, 1);
            s_posval[idx] = iprow[j];
        }
    }
    __syncthreads();

    const int npos = s_cnt;
    const int nneg = B - npos;

    float lsum = 0.0f;
    for (int k = tid; k < B; k += 256) {
        if (labels[k] != lb) {
            const float ipk = iprow[k] + ALPHA;   // fold alpha into negative side
            for (int jj = 0; jj < npos; ++jj) {
                float t = s_posval[jj] - ipk;
                t = fminf(fmaxf(t, -100.0f), 50.0f);
                lsum += log1pf(expf(t)) - t;      // softplus(-t), clipped form
            }
        }
    }

    // block reduction
    s_red[tid] = lsum;
    __syncthreads();
    for (int off = 128; off > 0; off >>= 1) {
        if (tid < off) s_red[tid] += s_red[tid + off];
        __syncthreads();
    }

    if (tid == 0) {
        const bool valid = (npos > 0) && (nneg > 0);
        const float denom = fmaxf((float)(npos * nneg), 1.0f);
        if (valid) {
            atomicAdd(&accum[0], s_red[0] / denom);  // sum of row means
            atomicAdd(&accum[1], 1.0f);              // valid row count
        }
    }
}

// ---------------------------------------------------------------------------
// Kernel 4: quantization SSE: sum((u - sign(u))^2); sign(0)=0 per jnp.sign
// ---------------------------------------------------------------------------
__global__ void __launch_bounds__(256)
quant_kernel(const float* __restrict__ u, float* __restrict__ accum) {
    __shared__ float s_red[256];
    const int tid = threadIdx.x;
    float lsum = 0.0f;
    for (int i = blockIdx.x * 256 + tid; i < B * NBIT; i += gridDim.x * 256) {
        float x = u[i];
        float s = (x > 0.0f) ? 1.0f : ((x < 0.0f) ? -1.0f : 0.0f);
        float d = x - s;
        lsum += d * d;
    }
    s_red[tid] = lsum;
    __syncthreads();
    for (int off = 128; off > 0; off >>= 1) {
        if (tid < off) s_red[tid] += s_red[tid + off];
        __syncthreads();
    }
    if (tid == 0) atomicAdd(&accum[2], s_red[0]);
}

// ---------------------------------------------------------------------------
// Kernel 5: finalize scalar loss
// ---------------------------------------------------------------------------
__global__ void finalize_kernel(const float* __restrict__ accum, float* __restrict__ out) {
    const float count = accum[1];
    const float loss1 = (count > 0.0f) ? (accum[0] / count) : 0.0f;
    const float loss2 = LMBD * (accum[2] / (float)(B * NBIT));
    out[0] = loss1 + loss2;
}

// ---------------------------------------------------------------------------
// Launcher. Workspace layout:
//   [0 .. B*B)              float  ip
//   [B*B .. B*B + B ints)   int    labels
//   then 3 floats           accum
// ---------------------------------------------------------------------------
extern "C" void kernel_launch(void* const* d_in, const int* in_sizes, int n_in,
                              void* d_out, int out_size, void* d_ws, size_t ws_size,
                              hipStream_t stream) {
    const float* u = (const float*)d_in[0];   // [512,64] f32
    const float* y = (const float*)d_in[1];   // [512,100] f32 one-hot
    // d_in[2] = ind (int64), unused by the loss

    float* ip     = (float*)d_ws;
    int*   labels = (int*)(ip + (size_t)B * B);
    float* accum  = (float*)(labels + B);
    float* out    = (float*)d_out;

    init_accum_kernel<<<1, 32, 0, stream>>>(accum);
    labels_kernel<<<(B + 255) / 256, 256, 0, stream>>>(y, labels);

    dim3 ggrid(B / 16, B / 16);   // 32 x 32 tiles, one wave each
    gemm_ip_wmma_kernel<<<ggrid, 32, 0, stream>>>(u, ip);

    triplet_row_kernel<<<B, 256, 0, stream>>>(ip, labels, accum);
    quant_kernel<<<64, 256, 0, stream>>>(u, accum);
    finalize_kernel<<<1, 1, 0, stream>>>(accum, out);
}